// EAAugmentedConv1d_69965017252030
// MI455X (gfx1250) — compile-verified
//
#include <hip/hip_runtime.h>
#include <hip/hip_bf16.h>

// ---------------------------------------------------------------------------
// EAAugmentedConv1d on gfx1250 (MI455X).
// All GEMMs use V_WMMA_F32_16X16X4_F32 (fp32 matrix pipe): the op is
// bandwidth-bound (~1 GB HBM traffic vs 28 GFLOP), so fp32 WMMA gives exact
// reference math while keeping the math off the VALU.
// ---------------------------------------------------------------------------

typedef __attribute__((ext_vector_type(2))) float v2f;
typedef __attribute__((ext_vector_type(8))) float v8f;

#define LSEQ   1024
#define CIN    512
#define OROWS  1536   // 2*DK + DV
#define NHEAD  8
#define DKH    64

static __device__ __forceinline__ v8f wmma_f32(v2f a, v2f b, v8f c) {
  // 8 args: (neg_a, A, neg_b, B, c_mod, C, reuse_a, reuse_b)
  return __builtin_amdgcn_wmma_f32_16x16x4_f32(false, a, false, b, (short)0, c,
                                               false, false);
}

// ---------------------------------------------------------------------------
// Kernel 1: qkv[b,o,l] = sum_c w[o,c] * x[b,c,l] + bias[o]
// One 16x16 tile per wave; K=512 -> 128 WMMA steps.
// ---------------------------------------------------------------------------
__global__ __launch_bounds__(256) void qkv_gemm_kernel(
    const float* __restrict__ x, const float* __restrict__ w,
    const float* __restrict__ bias, float* __restrict__ qkv) {
  const int wave = threadIdx.x >> 5;
  const int lane = threadIdx.x & 31;
  const int tile = blockIdx.x * 8 + wave;          // 4*96*64 tiles
  const int nt   = tile & 63;
  const int rest = tile >> 6;
  const int mt   = rest % 96;
  const int b    = rest / 96;
  const int m  = lane & 15;
  const int hi = lane >> 4;

  const float* __restrict__ wrow = w + (size_t)(mt * 16 + m) * CIN;
  const float* __restrict__ xcol =
      x + (size_t)b * CIN * LSEQ + (nt * 16 + m);

  v8f c = {0.f, 0.f, 0.f, 0.f, 0.f, 0.f, 0.f, 0.f};
  for (int kk = 0; kk < CIN; kk += 4) {
    const int kl = kk + 2 * hi;
    v2f a = *(const v2f*)(wrow + kl);              // A: row m, k=kl,kl+1
    v2f bf;
    bf.x = xcol[(size_t)(kl + 0) * LSEQ];          // B: col m, k=kl,kl+1
    bf.y = xcol[(size_t)(kl + 1) * LSEQ];
    c = wmma_f32(a, bf, c);
  }

  float* __restrict__ outp =
      qkv + (size_t)b * OROWS * LSEQ + (size_t)(mt * 16) * LSEQ + nt * 16;
#pragma unroll
  for (int r = 0; r < 8; ++r) {
    const int row = r + 8 * hi;
    outp[(size_t)row * LSEQ + m] = c[r] + bias[mt * 16 + row];
  }
}

// ---------------------------------------------------------------------------
// Kernel 2: att[b,h,i,j] = 0.5*(q.k + rel_to_abs) + 0.5*prev_att
// Per tile: shared scaled-q A fragment feeds 3 WMMAs/step (qk + 2 rel tiles
// covering key_rel rows [base-15, base+16]); rel diagonal gathered via LDS.
// ---------------------------------------------------------------------------
__global__ __launch_bounds__(256) void logits_kernel(
    const float* __restrict__ qkv, const float* __restrict__ key_rel,
    const float* __restrict__ prev_att, float* __restrict__ att) {
  __shared__ float sP[8][16 * 33];
  const int wave = threadIdx.x >> 5;
  const int lane = threadIdx.x & 31;
  int t = blockIdx.x * 8 + wave;                   // 4*8*64*64 tiles
  const int nt = t & 63; t >>= 6;
  const int mt = t & 63; t >>= 6;
  const int h  = t & 7;
  const int b  = t >> 3;
  const int n  = lane & 15;
  const int hi = lane >> 4;
  const int i0 = mt * 16, j0 = nt * 16;
  const int base = (LSEQ - 1) + j0 - i0;           // in [15, 2L-17]

  const float* __restrict__ qb = qkv + (size_t)b * OROWS * LSEQ;
  const float* __restrict__ qcol = qb + (size_t)(h * DKH) * LSEQ + i0 + n;
  const float* __restrict__ kcol =
      qb + (size_t)(512 + h * DKH) * LSEQ + j0 + n;
  const float* __restrict__ r0p = key_rel + (size_t)(base - 15 + n) * DKH;
  int r1row = base + 1 + n;                        // col 31 (unused) may OOB
  if (r1row > 2 * LSEQ - 2) r1row = 2 * LSEQ - 2;
  const float* __restrict__ r1p = key_rel + (size_t)r1row * DKH;

  v8f cqk = {0.f, 0.f, 0.f, 0.f, 0.f, 0.f, 0.f, 0.f};
  v8f cp0 = {0.f, 0.f, 0.f, 0.f, 0.f, 0.f, 0.f, 0.f};
  v8f cp1 = {0.f, 0.f, 0.f, 0.f, 0.f, 0.f, 0.f, 0.f};
  for (int kk = 0; kk < DKH; kk += 4) {
    const int kl = kk + 2 * hi;
    v2f a;                                          // scaled q (DKH^-0.5)
    a.x = qcol[(size_t)(kl + 0) * LSEQ] * 0.125f;
    a.y = qcol[(size_t)(kl + 1) * LSEQ] * 0.125f;
    v2f bk;
    bk.x = kcol[(size_t)(kl + 0) * LSEQ];
    bk.y = kcol[(size_t)(kl + 1) * LSEQ];
    v2f b0 = *(const v2f*)(r0p + kl);
    v2f b1 = *(const v2f*)(r1p + kl);
    cqk = wmma_f32(a, bk, cqk);
    cp0 = wmma_f32(a, b0, cp0);
    cp1 = wmma_f32(a, b1, cp1);
  }

  // Stage 16x32 rel product, gather skewed diagonal s = jj - ii + 15.
#pragma unroll
  for (int r = 0; r < 8; ++r) {
    const int ii = r + 8 * hi;
    sP[wave][ii * 33 + n]      = cp0[r];
    sP[wave][ii * 33 + 16 + n] = cp1[r];
  }
  __syncthreads();

  const size_t obase = (((size_t)(b * NHEAD + h) * LSEQ + i0) * LSEQ) + j0;
#pragma unroll
  for (int r = 0; r < 8; ++r) {
    const int ii  = r + 8 * hi;
    const float rel = sP[wave][ii * 33 + (n - ii + 15)];
    const size_t idx = obase + (size_t)ii * LSEQ + n;
    att[idx] = 0.5f * (cqk[r] + rel) + 0.5f * prev_att[idx];
  }
}

// ---------------------------------------------------------------------------
// Kernel 3: 3x3 NCHW conv (8->8 ch) + bias + leaky-ReLU + alpha-mix.
// LDS-tiled 32x32 spatial tile with halo, all 8 channels. Memory-bound.
// ---------------------------------------------------------------------------
__global__ __launch_bounds__(256) void conv_mix_kernel(
    const float* __restrict__ att, const float* __restrict__ cw,
    const float* __restrict__ cb, float* __restrict__ lg) {
  __shared__ float s_t[8][34][34];
  __shared__ float s_w[576];
  __shared__ float s_b[8];
  int bid = blockIdx.x;
  const int tj = bid & 31; bid >>= 5;
  const int ti = bid & 31; bid >>= 5;
  const int b  = bid;
  const int tid = threadIdx.x;
  const int ib = ti * 32 - 1, jb = tj * 32 - 1;

  for (int e = tid; e < 8 * 34 * 34; e += 256) {
    const int ci = e / 1156;
    const int rem = e - ci * 1156;
    const int li = rem / 34, lj = rem - li * 34;
    const int gi = ib + li, gj = jb + lj;
    float v = 0.f;
    if ((unsigned)gi < 1024u && (unsigned)gj < 1024u)
      v = att[((size_t)(b * 8 + ci) * LSEQ + gi) * LSEQ + gj];
    s_t[ci][li][lj] = v;
  }
  for (int e = tid; e < 576; e += 256) s_w[e] = cw[e];
  if (tid < 8) s_b[tid] = cb[tid];
  __syncthreads();

#pragma unroll
  for (int p = 0; p < 4; ++p) {
    const int px = tid + p * 256;
    const int pi = px >> 5, pj = px & 31;
    float acc[8];
#pragma unroll
    for (int co = 0; co < 8; ++co) acc[co] = s_b[co];
    for (int ci = 0; ci < 8; ++ci) {
      float nb[9];
#pragma unroll
      for (int kh = 0; kh < 3; ++kh)
#pragma unroll
        for (int kw = 0; kw < 3; ++kw)
          nb[kh * 3 + kw] = s_t[ci][pi + kh][pj + kw];
#pragma unroll
      for (int co = 0; co < 8; ++co) {
        const float* wp = s_w + co * 72 + ci * 9;
        float s = acc[co];
#pragma unroll
        for (int k = 0; k < 9; ++k) s += wp[k] * nb[k];
        acc[co] = s;
      }
    }
    const int gi = ti * 32 + pi, gj = tj * 32 + pj;
#pragma unroll
    for (int co = 0; co < 8; ++co) {
      float cv = acc[co];
      cv = (cv >= 0.f) ? cv : 0.01f * cv;                      // leaky ReLU
      const float lv = 0.5f * cv + 0.5f * s_t[co][pi + 1][pj + 1];
      lg[((size_t)(b * 8 + co) * LSEQ + gi) * LSEQ + gj] = lv;
    }
  }
}

// ---------------------------------------------------------------------------
// Kernel 4: row softmax (1024 wide), one wave32 per row, shfl reductions.
// ---------------------------------------------------------------------------
__global__ __launch_bounds__(256) void softmax_kernel(
    const float* __restrict__ lg, float* __restrict__ wts) {
  const int wave = threadIdx.x >> 5;
  const int lane = threadIdx.x & 31;
  const size_t row = (size_t)blockIdx.x * 8 + wave;
  const float* __restrict__ src = lg + row * LSEQ + lane;
  float v[32];
  float m = -3.402823466e38f;
#pragma unroll
  for (int k = 0; k < 32; ++k) {
    v[k] = src[k * 32];
    m = fmaxf(m, v[k]);
  }
#pragma unroll
  for (int off = 16; off >= 1; off >>= 1) m = fmaxf(m, __shfl_xor(m, off, 32));
  float s = 0.f;
#pragma unroll
  for (int k = 0; k < 32; ++k) {
    v[k] = expf(v[k] - m);
    s += v[k];
  }
#pragma unroll
  for (int off = 16; off >= 1; off >>= 1) s += __shfl_xor(s, off, 32);
  const float inv = 1.0f / s;
  float* __restrict__ dst = wts + row * LSEQ + lane;
#pragma unroll
  for (int k = 0; k < 32; ++k) dst[k * 32] = v[k] * inv;
}

// ---------------------------------------------------------------------------
// Kernel 5: out[b,h, l*64+d] = sum_m weights[b,h,l,m] * v[b,h,d,m]
// (reference's reshape is a flat reinterpret, so store (l,d) row-major)
// ---------------------------------------------------------------------------
__global__ __launch_bounds__(256) void attn_out_kernel(
    const float* __restrict__ wts, const float* __restrict__ qkv,
    float* __restrict__ out) {
  const int wave = threadIdx.x >> 5;
  const int lane = threadIdx.x & 31;
  int t = blockIdx.x * 8 + wave;                   // 4*8*64*4 tiles
  const int nt = t & 3;  t >>= 2;
  const int mt = t & 63; t >>= 6;
  const int h  = t & 7;
  const int b  = t >> 3;
  const int n  = lane & 15;
  const int hi = lane >> 4;
  const int i0 = mt * 16, j0 = nt * 16;

  const float* __restrict__ wrow =
      wts + ((size_t)(b * NHEAD + h) << 20) + (size_t)(i0 + n) * LSEQ;
  const float* __restrict__ vrow =
      qkv + (size_t)b * OROWS * LSEQ +
      (size_t)(1024 + h * DKH + j0 + n) * LSEQ;

  v8f c = {0.f, 0.f, 0.f, 0.f, 0.f, 0.f, 0.f, 0.f};
  for (int kk = 0; kk < LSEQ; kk += 4) {
    const int kl = kk + 2 * hi;
    v2f a  = *(const v2f*)(wrow + kl);
    v2f bf = *(const v2f*)(vrow + kl);
    c = wmma_f32(a, bf, c);
  }

  float* __restrict__ op =
      out + (size_t)(b * NHEAD + h) * (DKH * LSEQ) + (size_t)i0 * 64 + j0;
#pragma unroll
  for (int r = 0; r < 8; ++r) {
    const int ii = r + 8 * hi;
    op[ii * 64 + n] = c[r];
  }
}

// ---------------------------------------------------------------------------
extern "C" void kernel_launch(void* const* d_in, const int* in_sizes, int n_in,
                              void* d_out, int out_size, void* d_ws,
                              size_t ws_size, hipStream_t stream) {
  (void)in_sizes; (void)n_in; (void)out_size; (void)ws_size;
  const float* x      = (const float*)d_in[0];  // (4,512,1024)
  const float* prev   = (const float*)d_in[1];  // (4,8,1024,1024)
  const float* qkv_w  = (const float*)d_in[2];  // (1536,512)
  const float* qkv_b  = (const float*)d_in[3];  // (1536,)
  const float* keyrel = (const float*)d_in[4];  // (2047,64)
  const float* att_w  = (const float*)d_in[5];  // (8,8,3,3)
  const float* att_b  = (const float*)d_in[6];  // (8,)

  float* out        = (float*)d_out;                       // 4*512*1024
  float* logits_out = out + (size_t)4 * 512 * 1024;        // 4*8*1024*1024

  float* ws  = (float*)d_ws;
  float* qkv = ws;                                         // 4*1536*1024
  float* att = ws + (size_t)4 * OROWS * LSEQ;              // 4*8*1024*1024
  float* wts = att;  // att dead after conv_mix; reuse for softmax weights

  qkv_gemm_kernel<<<3072, 256, 0, stream>>>(x, qkv_w, qkv_b, qkv);
  logits_kernel<<<16384, 256, 0, stream>>>(qkv, keyrel, prev, att);
  conv_mix_kernel<<<4096, 256, 0, stream>>>(att, att_w, att_b, logits_out);
  softmax_kernel<<<4096, 256, 0, stream>>>(logits_out, wts);
  attn_out_kernel<<<1024, 256, 0, stream>>>(wts, qkv, out);
}